// GCNNet_68856915689565
// MI455X (gfx1250) — compile-verified
//
#include <hip/hip_runtime.h>

// GraphSAGE 4-layer forward for MI455X (gfx1250, wave32).
// Edge scatter is the roofline bottleneck (atomic-bound, L2-resident);
// dense linears use V_WMMA_F32_16X16X4_F32 with packed zero-padded B
// (no divergent loads in the k-loop) and a fused bias + row L2-norm +
// ReLU epilogue using ds_swizzle_b32 xor reductions and v_rsq_f32.

#define NN 100000
#define NE 1600000

typedef __attribute__((ext_vector_type(2))) float v2f;
typedef __attribute__((ext_vector_type(8))) float v8f;

__global__ void k_zero(float* __restrict__ p, int n) {
  int i = blockIdx.x * blockDim.x + threadIdx.x;
  if (i < n) p[i] = 0.0f;
}

__global__ void k_count(const int* __restrict__ dst, float* __restrict__ cnt, int e) {
  int i = blockIdx.x * blockDim.x + threadIdx.x;
  if (i < e) atomicAdd(&cnt[dst[i]], 1.0f);
}

__global__ void k_recip_clamp(float* __restrict__ p, int n) {
  int i = blockIdx.x * blockDim.x + threadIdx.x;
  if (i < n) p[i] = 1.0f / fmaxf(p[i], 1.0f);
}

// One thread per edge: s[dst] += h[src], C consecutive f32 atomics.
template <int C>
__global__ void k_scatter(const float* __restrict__ h, const int* __restrict__ src,
                          const int* __restrict__ dst, float* __restrict__ s, int e) {
  int i = blockIdx.x * blockDim.x + threadIdx.x;
  if (i >= e) return;
  const float* hr = h + (long long)src[i] * C;
  float* sr = s + (long long)dst[i] * C;
#pragma unroll
  for (int c = 0; c < C; ++c) atomicAdd(&sr[c], hr[c]);
}

// In-place mean: s[node][*] *= inv_degree[node]  (hoists the divide out of
// the WMMA loop so the A operand is a plain b64 load).
template <int C>
__global__ void k_scale(float* __restrict__ s, const float* __restrict__ inv, int n) {
  int i = blockIdx.x * blockDim.x + threadIdx.x;
  if (i < n) {
    float iv = inv[i];
    float* r = s + (size_t)i * C;
#pragma unroll
    for (int c = 0; c < C; ++c) r[c] *= iv;
  }
}

// Pre-pack B = [wl.T ; wr.T], zero-padded to 16 columns per n-tile, into the
// exact per-lane WMMA layout: entry t=(tile,j,lane) holds the v2f the lane
// feeds to V_WMMA_F32_16X16X4_F32 at k-step j. Also packs zero-padded bias.
// This removes all exec-mask-guarded loads from the hot linear kernel.
template <int CIN, int COUT>
__global__ void k_pack_b(const float* __restrict__ wl, const float* __restrict__ bl,
                         const float* __restrict__ wr, float* __restrict__ bpack,
                         float* __restrict__ biasp) {
  constexpr int KSTEPS = (2 * CIN) / 4;
  constexpr int NT = (COUT > 16) ? 2 : 1;
  int t = threadIdx.x;
  if (t < NT * KSTEPS * 32) {
    int tile = t / (KSTEPS * 32);
    int rem = t - tile * KSTEPS * 32;
    int j = rem >> 5;
    int lane = rem & 31;
    int n = (lane & 15) + tile * 16;
    int k0 = j * 4 + ((lane >> 4) << 1);
    float v0 = 0.0f, v1 = 0.0f;
    if (n < COUT) {
      int k1 = k0 + 1;
      v0 = (k0 < CIN) ? wl[n * CIN + k0] : wr[n * CIN + (k0 - CIN)];
      v1 = (k1 < CIN) ? wl[n * CIN + k1] : wr[n * CIN + (k1 - CIN)];
    }
    bpack[2 * t] = v0;
    bpack[2 * t + 1] = v1;
  }
  if (t < 32) biasp[t] = (t < COUT) ? bl[t] : 0.0f;
}

// ds_swizzle_b32 xor-swap add; PAT must be an immediate -> template parameter.
template <int PAT>
__device__ __forceinline__ float xor_add(float x) {
  return x + __int_as_float(__builtin_amdgcn_ds_swizzle(__float_as_int(x), PAT));
}

// Fused SAGE linear: out = relu( L2norm( [agg|x] @ [wl;wr].T + bl ) )
// One wave = one 16-node tile; K = 2*CIN stepped 4 at a time.
// A layout (ISA 16x4 f32): lanes 0-15 hold K={4j,4j+1}, lanes 16-31 K={4j+2,4j+3};
// since CIN is even, each K-pair lies entirely in agg or in x -> pointer
// select + one unconditional global_load_b64 (no saveexec).
// D layout (ISA 16x16 f32): lanes 0-15 -> rows 0-7 (vgpr=row), lanes 16-31 -> rows 8-15.
template <int CIN, int COUT>
__global__ __launch_bounds__(64) void k_sage_linear(
    const float* __restrict__ agg,    // pre-scaled sums (= mean agg), [N, CIN]
    const float* __restrict__ hin,    // own features, [N, CIN]
    const float* __restrict__ bpack,  // packed B (v2f per (tile,j,lane))
    const float* __restrict__ biasp,  // [32] zero-padded bias
    float* __restrict__ out) {
  constexpr int KSTEPS = (2 * CIN) / 4;
  constexpr bool TWO = (COUT > 16);

  const int lane = threadIdx.x & 31;
  const int wave = threadIdx.x >> 5;
  const int tile = blockIdx.x * 2 + wave;  // NN/16 tiles, 2 waves/block
  const int m = lane & 15;
  const int khalf = (lane >> 4) << 1;
  const int node = tile * 16 + m;
  const int ncol = lane & 15;

  const float* arow = agg + node * CIN;
  const float* hrow = hin + node * CIN;
  const v2f* bpk = (const v2f*)bpack;

  v8f acc0 = {};
  v8f acc1 = {};

#pragma unroll
  for (int j = 0; j < KSTEPS; ++j) {
    const int k0 = j * 4 + khalf;
    const float* ap = (k0 < CIN) ? (arow + k0) : (hrow + (k0 - CIN));
    v2f a = *(const v2f*)ap;
    v2f b0 = bpk[j * 32 + lane];
    acc0 = __builtin_amdgcn_wmma_f32_16x16x4_f32(false, a, false, b0, (short)0,
                                                 acc0, false, false);
    if constexpr (TWO) {
      v2f b1 = bpk[(KSTEPS + j) * 32 + lane];
      acc1 = __builtin_amdgcn_wmma_f32_16x16x4_f32(false, a, false, b1, (short)0,
                                                   acc1, false, false);
    }
  }

  // Epilogue: bias, row-wise L2 norm (16 lanes/row via ds_swizzle xor), ReLU.
  const int rowoff = (lane >> 4) * 8;
  const float bias0 = biasp[ncol];  // zero-padded -> padded cols stay 0
  float bias1 = 0.0f;
  if constexpr (TWO) bias1 = biasp[ncol + 16];

#pragma unroll
  for (int v = 0; v < 8; ++v) {
    float d0 = acc0[v] + bias0;
    float d1 = 0.0f;
    if constexpr (TWO) d1 = acc1[v] + bias1;
    float ss = d0 * d0 + d1 * d1;
    ss = xor_add<(1 << 10) | 0x1f>(ss);  // ds_swizzle_b32 xor-swaps, group of 32
    ss = xor_add<(2 << 10) | 0x1f>(ss);
    ss = xor_add<(4 << 10) | 0x1f>(ss);
    ss = xor_add<(8 << 10) | 0x1f>(ss);  // masks < 16: stays within each half-wave
    // 1/max(sqrt(ss),1e-12) == rsqrt(max(ss,1e-24)) -> single v_rsq_f32
    const float rinv = __builtin_amdgcn_rsqf(fmaxf(ss, 1e-24f));
    const int row = tile * 16 + rowoff + v;
    if constexpr (COUT >= 16) {
      out[row * COUT + ncol] = fmaxf(d0 * rinv, 0.0f);
    } else {
      if (ncol < COUT) out[row * COUT + ncol] = fmaxf(d0 * rinv, 0.0f);
    }
    if constexpr (TWO) out[row * COUT + ncol + 16] = fmaxf(d1 * rinv, 0.0f);
  }
}

// Per-channel sum and sum-of-squares; LDS partials, then global atomics.
template <int C>
__global__ void k_bnstats(const float* __restrict__ h, float* __restrict__ stats, int n) {
  __shared__ float sh[2 * C];
  if (threadIdx.x < 2 * C) sh[threadIdx.x] = 0.0f;
  __syncthreads();
  float ls[C], lq[C];
#pragma unroll
  for (int c = 0; c < C; ++c) { ls[c] = 0.0f; lq[c] = 0.0f; }
  for (int i = blockIdx.x * blockDim.x + threadIdx.x; i < n; i += gridDim.x * blockDim.x) {
    const float* r = h + i * C;
#pragma unroll
    for (int c = 0; c < C; ++c) {
      float x = r[c];
      ls[c] += x;
      lq[c] += x * x;
    }
  }
#pragma unroll
  for (int c = 0; c < C; ++c) {
    atomicAdd(&sh[c], ls[c]);
    atomicAdd(&sh[C + c], lq[c]);
  }
  __syncthreads();
  if (threadIdx.x < 2 * C) atomicAdd(&stats[threadIdx.x], sh[threadIdx.x]);
}

__global__ void k_bnfinal(const float* __restrict__ stats, const float* __restrict__ g,
                          const float* __restrict__ be, float* __restrict__ sshift,
                          int C, float invN) {
  int c = threadIdx.x;
  if (c < C) {
    float m = stats[c] * invN;
    float var = fmaxf(stats[C + c] * invN - m * m, 0.0f);  // biased variance
    float sc = g[c] * __builtin_amdgcn_rsqf(var + 1e-5f);
    sshift[c] = sc;
    sshift[C + c] = be[c] - m * sc;
  }
}

template <int C>
__global__ void k_bnapply(float* __restrict__ h, const float* __restrict__ sshift, int total) {
  int i = blockIdx.x * blockDim.x + threadIdx.x;
  if (i < total) {
    int c = i % C;
    h[i] = h[i] * sshift[c] + sshift[C + c];
  }
}

extern "C" void kernel_launch(void* const* d_in, const int* in_sizes, int n_in,
                              void* d_out, int out_size, void* d_ws, size_t ws_size,
                              hipStream_t stream) {
  (void)in_sizes; (void)n_in; (void)out_size; (void)ws_size;
  const float* x = (const float*)d_in[0];
  const int* ei = (const int*)d_in[1];  // [2, E] int32 (JAX x64 disabled)
  const int* src = ei;
  const int* dst = ei + NE;
  const float* w1l = (const float*)d_in[2];
  const float* b1l = (const float*)d_in[3];
  const float* w1r = (const float*)d_in[4];
  const float* w2l = (const float*)d_in[5];
  const float* b2l = (const float*)d_in[6];
  const float* w2r = (const float*)d_in[7];
  const float* w3l = (const float*)d_in[8];
  const float* b3l = (const float*)d_in[9];
  const float* w3r = (const float*)d_in[10];
  const float* w4l = (const float*)d_in[11];
  const float* b4l = (const float*)d_in[12];
  const float* w4r = (const float*)d_in[13];
  const float* g1 = (const float*)d_in[14];
  const float* be1 = (const float*)d_in[15];
  const float* g2 = (const float*)d_in[16];
  const float* be2 = (const float*)d_in[17];
  const float* g3 = (const float*)d_in[18];
  const float* be3 = (const float*)d_in[19];

  float* ws = (float*)d_ws;
  float* inv = ws;                          // N
  float* s = ws + (size_t)NN;               // N*16 (max Cin)
  float* hA = ws + (size_t)17 * NN;         // N*16
  float* hB = ws + (size_t)33 * NN;         // N*16
  float* stats = ws + (size_t)49 * NN;      // 64
  float* sshift = stats + 64;               // 64
  float* biasp = stats + 128;               // 32
  float* bpack = stats + 160;               // <= 1024 (2 tiles * 8 ksteps * 32 lanes * 2)
  float* outF = (float*)d_out;              // [N, 32]

  const int TB = 256;
  const int gE = (NE + TB - 1) / TB;
  const int gN = (NN + TB - 1) / TB;
  const int gLin = NN / 32;  // 6250 tiles, 2 waves (tiles) per 64-thread block
  const float invN = 1.0f / (float)NN;

  // Degree counts once (reused by all 4 layers), then reciprocal-clamp.
  k_zero<<<gN, TB, 0, stream>>>(inv, NN);
  k_count<<<gE, TB, 0, stream>>>(dst, inv, NE);
  k_recip_clamp<<<gN, TB, 0, stream>>>(inv, NN);

  // ---- Layer 1: 4 -> 6 ----
  k_zero<<<(NN * 4 + TB - 1) / TB, TB, 0, stream>>>(s, NN * 4);
  k_scatter<4><<<gE, TB, 0, stream>>>(x, src, dst, s, NE);
  k_scale<4><<<gN, TB, 0, stream>>>(s, inv, NN);
  k_pack_b<4, 6><<<1, 512, 0, stream>>>(w1l, b1l, w1r, bpack, biasp);
  k_sage_linear<4, 6><<<gLin, 64, 0, stream>>>(s, x, bpack, biasp, hA);
  k_zero<<<1, 64, 0, stream>>>(stats, 12);
  k_bnstats<6><<<256, 256, 0, stream>>>(hA, stats, NN);
  k_bnfinal<<<1, 32, 0, stream>>>(stats, g1, be1, sshift, 6, invN);
  k_bnapply<6><<<(NN * 6 + TB - 1) / TB, TB, 0, stream>>>(hA, sshift, NN * 6);

  // ---- Layer 2: 6 -> 8 ----
  k_zero<<<(NN * 6 + TB - 1) / TB, TB, 0, stream>>>(s, NN * 6);
  k_scatter<6><<<gE, TB, 0, stream>>>(hA, src, dst, s, NE);
  k_scale<6><<<gN, TB, 0, stream>>>(s, inv, NN);
  k_pack_b<6, 8><<<1, 512, 0, stream>>>(w2l, b2l, w2r, bpack, biasp);
  k_sage_linear<6, 8><<<gLin, 64, 0, stream>>>(s, hA, bpack, biasp, hB);
  k_zero<<<1, 64, 0, stream>>>(stats, 16);
  k_bnstats<8><<<256, 256, 0, stream>>>(hB, stats, NN);
  k_bnfinal<<<1, 32, 0, stream>>>(stats, g2, be2, sshift, 8, invN);
  k_bnapply<8><<<(NN * 8 + TB - 1) / TB, TB, 0, stream>>>(hB, sshift, NN * 8);

  // ---- Layer 3: 8 -> 16 ----
  k_zero<<<(NN * 8 + TB - 1) / TB, TB, 0, stream>>>(s, NN * 8);
  k_scatter<8><<<gE, TB, 0, stream>>>(hB, src, dst, s, NE);
  k_scale<8><<<gN, TB, 0, stream>>>(s, inv, NN);
  k_pack_b<8, 16><<<1, 512, 0, stream>>>(w3l, b3l, w3r, bpack, biasp);
  k_sage_linear<8, 16><<<gLin, 64, 0, stream>>>(s, hB, bpack, biasp, hA);
  k_zero<<<1, 64, 0, stream>>>(stats, 32);
  k_bnstats<16><<<256, 256, 0, stream>>>(hA, stats, NN);
  k_bnfinal<<<1, 32, 0, stream>>>(stats, g3, be3, sshift, 16, invN);
  k_bnapply<16><<<(NN * 16 + TB - 1) / TB, TB, 0, stream>>>(hA, sshift, NN * 16);

  // ---- Layer 4: 16 -> 32 (ReLU only, straight to d_out) ----
  k_zero<<<(NN * 16 + TB - 1) / TB, TB, 0, stream>>>(s, NN * 16);
  k_scatter<16><<<gE, TB, 0, stream>>>(hA, src, dst, s, NE);
  k_scale<16><<<gN, TB, 0, stream>>>(s, inv, NN);
  k_pack_b<16, 32><<<1, 512, 0, stream>>>(w4l, b4l, w4r, bpack, biasp);
  k_sage_linear<16, 32><<<gLin, 64, 0, stream>>>(s, hA, bpack, biasp, outF);
}